// Block_21251498181151
// MI455X (gfx1250) — compile-verified
//
#include <hip/hip_runtime.h>
#include <math.h>

// ---------------------------------------------------------------------------
// CDNA5 (gfx1250) equivariant message-passing kernel.
// 16 edges per wave32; every channel<->rank projection is a chain of
// V_WMMA_F32_16X16X4_F32 (rows = rank/out-channel, cols = edge, K = channels).
// The radial factor phi = rad (x) rdir (x) rdir factorizes, so each TP's right
// projection is one 16x16 "w" tile (Wr x rad, 2 WMMAs) and the spatial
// contractions are elementwise-in-rank in the WMMA C/D register layout.
// D-layout -> B-layout conversion for the Wo projection uses ds_bpermute
// half-swaps. Edges beyond the cutoff (relu(1-|r|^2/R0)==0) contribute exactly
// zero and are removed by a compaction pre-pass (~3x work reduction).
// Round-3 change: the 72 gathered x_d B-operand floats per lane are staged in
// registers ONCE per edge-group and shared by the four rank-2-left TPs
// (tp_220/tp_211/tp_202/tp_222) instead of being re-loaded per TP (-216 vmem
// instructions per group), while keeping the low-pressure per-cell scatter
// structure from round 2.
// ---------------------------------------------------------------------------

typedef float v2f __attribute__((ext_vector_type(2)));
typedef float v8f __attribute__((ext_vector_type(8)));

struct TPParams { const float* p[27]; };
// param order (setup_inputs insertion order):
//  i: 0 tp_000 | 1 tp_101 | 2 tp_202 | 3 tp_011 | 4 tp_110 | 5 tp_112
//     6 tp_211 | 7 tp_220 | 8 tp_222 ; p[3i]=Wl, p[3i+1]=Wr, p[3i+2]=Wo

__device__ __forceinline__ v8f wmma4(v2f a, v2f b, v8f c) {
  // D = A(16x4) * B(4x16) + C, fp32
  return __builtin_amdgcn_wmma_f32_16x16x4_f32(false, a, false, b, (short)0, c,
                                               false, false);
}

__device__ __forceinline__ float swap_half(float x, int swidx) {
  // exchange lanes 0-15 <-> 16-31
  return __int_as_float(__builtin_amdgcn_ds_bpermute(swidx, __float_as_int(x)));
}

// Left projection: D[rank, edge] = Wl(16xK) * X(K x 16edges).
// Wl row-major (16,K). xb points at feature 0 of this lane's edge; S = stride
// between consecutive features (1 for x_a, 3 for x_v comp).
template <int K, int S>
__device__ __forceinline__ v8f lproj(const float* __restrict__ Wl,
                                     const float* __restrict__ xb,
                                     int m, int half) {
  v8f acc = {};
#pragma unroll
  for (int t = 0; t < K / 4; ++t) {
    const int c = 4 * t + 2 * half;        // A: lane m holds Wl[m][c], Wl[m][c+1]
    v2f a = *(const v2f*)(Wl + m * K + c); // B: lane e holds X[c][e], X[c+1][e]
    v2f b;
    if constexpr (S == 1) {
      b = *(const v2f*)(xb + c);
    } else {
      b.x = xb[c * S];
      b.y = xb[(c + 1) * S];
    }
    acc = wmma4(a, b, acc);
  }
  return acc;
}

// Left projection with pre-staged B-operand registers (K = 16, x_d cells).
__device__ __forceinline__ v8f lproj16c(const float* __restrict__ Wl,
                                        const v2f bc[4], int m, int half) {
  v8f acc = {};
#pragma unroll
  for (int t = 0; t < 4; ++t) {
    v2f a = *(const v2f*)(Wl + m * 16 + 4 * t + 2 * half);
    acc = wmma4(a, bc[t], acc);
  }
  return acc;
}

// w[rank, edge] = Wr(16x8) * rad(8 x 16edges)
__device__ __forceinline__ v8f wproj(const float* __restrict__ Wr,
                                     const v2f radB0, const v2f radB1,
                                     int m, int half) {
  v8f acc = {};
  v2f a0 = *(const v2f*)(Wr + m * 8 + 2 * half);
  acc = wmma4(a0, radB0, acc);
  v2f a1 = *(const v2f*)(Wr + m * 8 + 4 + 2 * half);
  acc = wmma4(a1, radB1, acc);
  return acc;
}

struct Bregs { v2f b[4]; };

// Convert y (C/D layout: reg v = ranks {v, v+8}) into B-operand k-step regs
// (k-step t needs ranks {4t,4t+2} in reg0 halves and {4t+1,4t+3} in reg1).
__device__ __forceinline__ Bregs yconv(v8f y, int swidx, bool lo) {
  float sw[8];
#pragma unroll
  for (int r = 0; r < 8; ++r) sw[r] = swap_half(y[r], swidx);
  Bregs o;
  o.b[0].x = lo ? y[0]  : sw[2];  o.b[0].y = lo ? y[1]  : sw[3]; // ranks 0-3
  o.b[1].x = lo ? y[4]  : sw[6];  o.b[1].y = lo ? y[5]  : sw[7]; // ranks 4-7
  o.b[2].x = lo ? sw[0] : y[2];   o.b[2].y = lo ? sw[1] : y[3];  // ranks 8-11
  o.b[3].x = lo ? sw[4] : y[6];   o.b[3].y = lo ? sw[5] : y[7];  // ranks 12-15
  return o;
}

// Out projection: acc[tile] += Wo(do x 16) * y(16 x 16edges), NT = do/16 tiles
template <int NT>
__device__ __forceinline__ void oproj(const float* __restrict__ Wo,
                                      const Bregs& B, v8f* acc,
                                      int m, int half) {
#pragma unroll
  for (int tt = 0; tt < NT; ++tt) {
    v8f c = acc[tt];
#pragma unroll
    for (int t = 0; t < 4; ++t) {
      v2f a = *(const v2f*)(Wo + (tt * 16 + m) * 16 + 4 * t + 2 * half);
      c = wmma4(a, B.b[t], c);
    }
    acc[tt] = c;
  }
}

// ---------------------------------------------------------------------------
// Pass 1: compact edges with non-zero radial envelope (x_sq < 1)
// ---------------------------------------------------------------------------
__global__ void compact_kernel(const float* __restrict__ r_ij, int E,
                               int* __restrict__ cnt, int* __restrict__ act) {
  int i = blockIdx.x * blockDim.x + threadIdx.x;
  if (i >= E) return;
  float rx = r_ij[3 * (size_t)i + 0];
  float ry = r_ij[3 * (size_t)i + 1];
  float rz = r_ij[3 * (size_t)i + 2];
  float xsq = (rx * rx + ry * ry + rz * rz) * (1.0f / 6.0f);
  if (xsq < 1.0f) {
    int k = atomicAdd(cnt, 1);
    act[k] = i;
  }
}

// ---------------------------------------------------------------------------
// Pass 2: per-edge tensor products + scatter
// ---------------------------------------------------------------------------
__global__ __launch_bounds__(256) void msg_kernel(
    const float* __restrict__ r_ij, const float* __restrict__ x_a,
    const float* __restrict__ x_v, const float* __restrict__ x_d,
    TPParams P, const int* __restrict__ src, const int* __restrict__ dst,
    const int* __restrict__ cnt, const int* __restrict__ act, int E,
    float* __restrict__ out_a, float* __restrict__ out_v,
    float* __restrict__ out_d) {
  const int lane = threadIdx.x & 31;
  const int half = lane >> 4;
  const bool lo = (half == 0);
  const int m = lane & 15;            // edge-in-group for B/C/D; row for A
  const int swidx = ((lane ^ 16) << 2);

  const int wpb = blockDim.x >> 5;
  const int gwave = blockIdx.x * wpb + (threadIdx.x >> 5);
  const int nwav = gridDim.x * wpb;

  const int nact = cnt ? cnt[0] : E;  // wave-uniform scalar load
  const int ngrp = (nact + 15) >> 4;

  for (int g = gwave; g < ngrp; g += nwav) {
    const int ei = g * 16 + m;
    const bool valid = (ei < nact);
    const int ci = valid ? ei : (nact - 1);
    const int e = act ? act[ci] : ci;

    // ---- per-edge scalars (both half-waves compute the same edge) ----
    const float rx = r_ij[3 * (size_t)e + 0];
    const float ry = r_ij[3 * (size_t)e + 1];
    const float rz = r_ij[3 * (size_t)e + 2];
    const float xsq = (rx * rx + ry * ry + rz * rz) * (1.0f / 6.0f);
    const float s = sqrtf(xsq);
    const float env = fmaxf(0.0f, 1.0f - xsq);
    float rad[8];
    {
      const float c1 = cosf(3.14159265358979323846f * s);
      float cm2 = 1.0f, cm1 = c1;
      rad[0] = env;
      rad[1] = c1 * env;
#pragma unroll
      for (int n = 2; n < 8; ++n) {
        float cn = 2.0f * c1 * cm1 - cm2;
        cm2 = cm1; cm1 = cn;
        rad[n] = cn * env;
      }
    }
    const float vx = rx * (17.0f / 6.0f);
    const float vy = ry * (17.0f / 6.0f);
    const float vz = rz * (17.0f / 6.0f);
    const float nn = sqrtf(vx * vx + vy * vy + vz * vz + 1e-12f);
    const float tsc = tanhf(0.5f * nn) * __builtin_amdgcn_rcpf(nn);
    const float u[3] = {vx * tsc, vy * tsc, vz * tsc};

    // rad in B-operand layout for the w-projection (constant indices only)
    v2f radB0, radB1;
    radB0.x = lo ? rad[0] : rad[2];  radB0.y = lo ? rad[1] : rad[3];
    radB1.x = lo ? rad[4] : rad[6];  radB1.y = lo ? rad[5] : rad[7];

    const int nj = dst[e];
    const int ns = src[e];
    const float* xa  = x_a + (size_t)nj * 64;
    const float* xv  = x_v + (size_t)nj * 96;    // (32,3)
    const float* xdp = x_d + (size_t)nj * 144;   // (16,3,3)

    // ---- stage gathered x_d B-operands once; reused by 4 TPs ----
    // cell (i,j): 16 channels at stride 9; k-step t needs channels
    // {4t+2h, 4t+2h+1} of this lane's edge.
    v2f xdB[9][4];
#pragma unroll
    for (int cell = 0; cell < 9; ++cell) {
#pragma unroll
      for (int t = 0; t < 4; ++t) {
        const int c = 4 * t + 2 * half;
        xdB[cell][t].x = xdp[c * 9 + cell];
        xdB[cell][t].y = xdp[(c + 1) * 9 + cell];
      }
    }

    // ================= psi_a : tp_000 + tp_110 + tp_220 =================
    {
      v8f pa[4] = {};
      { // tp_000: y[r] = l_a[r] * w[r]
        v8f w = wproj(P.p[1], radB0, radB1, m, half);
        v8f l = lproj<64, 1>(P.p[0], xa, m, half);
        Bregs B = yconv(l * w, swidx, lo);
        oproj<4>(P.p[2], B, pa, m, half);
      }
      { // tp_110: y[r] = w[r] * sum_i l_i[r] u_i
        v8f w = wproj(P.p[13], radB0, radB1, m, half);
        v8f acc = {};
#pragma unroll
        for (int i = 0; i < 3; ++i) {
          acc += lproj<32, 3>(P.p[12], xv + i, m, half) * u[i];
        }
        Bregs B = yconv(acc * w, swidx, lo);
        oproj<4>(P.p[14], B, pa, m, half);
      }
      { // tp_220: y[r] = w[r] * sum_ij l_ij[r] u_i u_j
        v8f w = wproj(P.p[22], radB0, radB1, m, half);
        v8f acc = {};
#pragma unroll
        for (int i = 0; i < 3; ++i) {
#pragma unroll
          for (int j = 0; j < 3; ++j) {
            acc += lproj16c(P.p[21], xdB[3 * i + j], m, half) * (u[i] * u[j]);
          }
        }
        Bregs B = yconv(acc * w, swidx, lo);
        oproj<4>(P.p[23], B, pa, m, half);
      }
      if (valid) {
        float* oa = out_a + (size_t)ns * 64;
#pragma unroll
        for (int tt = 0; tt < 4; ++tt) {
#pragma unroll
          for (int v = 0; v < 8; ++v) {
            atomicAdd(oa + tt * 16 + v + 8 * half, pa[tt][v]);
          }
        }
      }
    }

    // ================= psi_v : tp_011 + tp_101 + tp_211 =================
    // Scatter per spatial component i so only 2 tiles are ever live.
    {
      v8f w3 = wproj(P.p[10], radB0, radB1, m, half);
      v8f lw3 = lproj<64, 1>(P.p[9], xa, m, half) * w3;   // tp_011 shared part
      v8f w1 = wproj(P.p[4], radB0, radB1, m, half);
      v8f w6 = wproj(P.p[19], radB0, radB1, m, half);
#pragma unroll
      for (int i = 0; i < 3; ++i) {
        v8f pv[2] = {};
        { // tp_011: y_i[r] = l_a[r] * w[r] * u_i
          Bregs B = yconv(lw3 * u[i], swidx, lo);
          oproj<2>(P.p[11], B, pv, m, half);
        }
        { // tp_101: y_i[r] = l_i[r] * w[r]
          v8f y = lproj<32, 3>(P.p[3], xv + i, m, half) * w1;
          Bregs B = yconv(y, swidx, lo);
          oproj<2>(P.p[5], B, pv, m, half);
        }
        { // tp_211: y_i[r] = w[r] * sum_j l_ij[r] u_j
          v8f acc = {};
#pragma unroll
          for (int j = 0; j < 3; ++j) {
            acc += lproj16c(P.p[18], xdB[3 * i + j], m, half) * u[j];
          }
          Bregs B = yconv(acc * w6, swidx, lo);
          oproj<2>(P.p[20], B, pv, m, half);
        }
        if (valid) {
          float* ov = out_v + (size_t)ns * 96;
#pragma unroll
          for (int tt = 0; tt < 2; ++tt) {
#pragma unroll
            for (int v = 0; v < 8; ++v) {
              atomicAdd(ov + (tt * 16 + v + 8 * half) * 3 + i, pv[tt][v]);
            }
          }
        }
      }
    }

    // ================= psi_d : tp_202 + tp_112 + tp_222 =================
    // Scatter per (i,k) cell so only 1 tile is ever live; hoist per-i pieces.
    {
      v8f w2 = wproj(P.p[7], radB0, radB1, m, half);
      v8f w5 = wproj(P.p[16], radB0, radB1, m, half);
      v8f w8 = wproj(P.p[25], radB0, radB1, m, half);
#pragma unroll
      for (int i = 0; i < 3; ++i) {
        // tp_112 shared part: l_i[r] * w[r]
        v8f lw5 = lproj<32, 3>(P.p[15], xv + i, m, half) * w5;
        // tp_222 shared part: w[r] * sum_j l_ij[r] u_j
        v8f acc8 = {};
#pragma unroll
        for (int j = 0; j < 3; ++j) {
          acc8 += lproj16c(P.p[24], xdB[3 * i + j], m, half) * u[j];
        }
        v8f aw8 = acc8 * w8;
#pragma unroll
        for (int k = 0; k < 3; ++k) {
          v8f pd[1] = {};
          { // tp_202: y_ik[r] = l_ik[r] * w[r]
            v8f y = lproj16c(P.p[6], xdB[3 * i + k], m, half) * w2;
            Bregs B = yconv(y, swidx, lo);
            oproj<1>(P.p[8], B, pd, m, half);
          }
          { // tp_112: y_ik[r] = l_i[r] * w[r] * u_k
            Bregs B = yconv(lw5 * u[k], swidx, lo);
            oproj<1>(P.p[17], B, pd, m, half);
          }
          { // tp_222: y_ik[r] = w[r] * u_k * sum_j l_ij[r] u_j
            Bregs B = yconv(aw8 * u[k], swidx, lo);
            oproj<1>(P.p[26], B, pd, m, half);
          }
          if (valid) {
            float* od = out_d + (size_t)ns * 144;
#pragma unroll
            for (int v = 0; v < 8; ++v) {
              atomicAdd(od + (v + 8 * half) * 9 + 3 * i + k, pd[0][v]);
            }
          }
        }
      }
    }
  }
}

// ---------------------------------------------------------------------------
extern "C" void kernel_launch(void* const* d_in, const int* in_sizes, int n_in,
                              void* d_out, int out_size, void* d_ws,
                              size_t ws_size, hipStream_t stream) {
  const float* r_ij = (const float*)d_in[0];
  const float* x_a = (const float*)d_in[1];
  const float* x_v = (const float*)d_in[2];
  const float* x_d = (const float*)d_in[3];
  TPParams P;
  for (int i = 0; i < 27; ++i) P.p[i] = (const float*)d_in[4 + i];
  const int* src = (const int*)d_in[n_in - 2];
  const int* dst = (const int*)d_in[n_in - 1];
  const int E = in_sizes[0] / 3;
  const int N = in_sizes[1] / 64;

  float* out = (float*)d_out;
  float* out_a = out;
  float* out_v = out + (size_t)N * 64;
  float* out_d = out + (size_t)N * 64 + (size_t)N * 96;

  hipMemsetAsync(d_out, 0, (size_t)out_size * sizeof(float), stream);

  int* cnt = nullptr;
  int* act = nullptr;
  const bool useCompact = (ws_size >= (size_t)E * sizeof(int) + 16);
  if (useCompact) {
    cnt = (int*)d_ws;
    act = (int*)d_ws + 4;  // 16-byte aligned offset
    hipMemsetAsync(d_ws, 0, 16, stream);
    compact_kernel<<<(E + 255) / 256, 256, 0, stream>>>(r_ij, E, cnt, act);
  }

  const int maxGroups = (E + 15) / 16;
  const int wavesPerBlock = 8;  // 256 threads = 8 wave32
  const int blocks = (maxGroups + wavesPerBlock - 1) / wavesPerBlock;
  msg_kernel<<<blocks, 256, 0, stream>>>(r_ij, x_a, x_v, x_d, P, src, dst, cnt,
                                         act, E, out_a, out_v, out_d);
}